// SkipConnector_62302795596004
// MI455X (gfx1250) — compile-verified
//
#include <hip/hip_runtime.h>
#include <cstddef>

// SkipConnector resample: out[b, s*16+c, d, h, w] = trilinear sample of a
// volume that's zero except plane w==48 where x[b,s*16+c,:,:] sits.
// Store-bound: 226 MB fp32 out (~10us @ 23.3 TB/s). 4 voxels/thread with
// b128 NT stores to keep instruction issue well below the store roofline.
// Grid einsum theta(3x4) x base(4x16) done with V_WMMA_F32_16X16X4_F32.

typedef __attribute__((ext_vector_type(2))) float v2f;
typedef __attribute__((ext_vector_type(4))) float v4f;
typedef __attribute__((ext_vector_type(8))) float v8f;

#define SPAT 96
#define XSL  (SPAT * SPAT)          // 9216 floats per channel slice of x
#define VOX  (SPAT * SPAT * SPAT)   // 884736 voxels
#define NCH  16                     // channels per slice chunk (C/N_SLICES)
#define THREADS 256
#define WPT  4                      // voxels along w per thread
#define QROW (SPAT / WPT)           // 24 quads per row
#define NQ   (VOX / WPT)            // 221184 quads total

__global__ __launch_bounds__(THREADS)
void skip_sample_kernel(const float* __restrict__ x,
                        const float* __restrict__ affs,
                        float* __restrict__ out)
{
    __shared__ float th[12];             // theta rows 0..2, cols 0..3

    const int sb = blockIdx.y;           // 0..3 : sb = s*2 + b  (B==2)
    const int s  = sb >> 1;
    const int b  = sb & 1;

    // ---- thread 0: theta = inv(aff * diag(1/zooms, 1))[:3,:] ----
    if (threadIdx.x == 0) {
        const float* A = affs + (size_t)sb * 16;
        float a[12];
        #pragma unroll
        for (int j = 0; j < 3; ++j) {
            float z = sqrtf(A[0*4+j]*A[0*4+j] + A[1*4+j]*A[1*4+j] + A[2*4+j]*A[2*4+j]);
            float rz = 1.0f / z;
            a[0*4+j] = A[0*4+j] * rz;
            a[1*4+j] = A[1*4+j] * rz;
            a[2*4+j] = A[2*4+j] * rz;
        }
        a[3] = A[3]; a[7] = A[7]; a[11] = A[11];   // translation col (scale 1)
        float m00=a[0], m01=a[1], m02=a[2];
        float m10=a[4], m11=a[5], m12=a[6];
        float m20=a[8], m21=a[9], m22=a[10];
        float c00 =  m11*m22 - m12*m21;
        float c01 = -(m10*m22 - m12*m20);
        float c02 =  m10*m21 - m11*m20;
        float invdet = 1.0f / (m00*c00 + m01*c01 + m02*c02);
        float i00 = c00*invdet;
        float i01 = (m02*m21 - m01*m22)*invdet;
        float i02 = (m01*m12 - m02*m11)*invdet;
        float i10 = c01*invdet;
        float i11 = (m00*m22 - m02*m20)*invdet;
        float i12 = (m02*m10 - m00*m12)*invdet;
        float i20 = c02*invdet;
        float i21 = (m01*m20 - m00*m21)*invdet;
        float i22 = (m00*m11 - m01*m10)*invdet;
        float tx = a[3], ty = a[7], tz = a[11];
        th[0]=i00; th[1]=i01; th[2] =i02; th[3] = -(i00*tx + i01*ty + i02*tz);
        th[4]=i10; th[5]=i11; th[6] =i12; th[7] = -(i10*tx + i11*ty + i12*tz);
        th[8]=i20; th[9]=i21; th[10]=i22; th[11]= -(i20*tx + i21*ty + i22*tz);
    }
    __syncthreads();

    const int q  = blockIdx.x * THREADS + threadIdx.x;   // quad index
    const int wq = q % QROW;
    const int r  = q / QROW;                             // row = d*96 + h
    const int h  = r % SPAT;
    const int d  = r / SPAT;
    const int w  = wq * WPT;                             // base voxel w

    const int  lane = threadIdx.x & 31;
    const int  row  = lane & 15;
    const int  kb   = (lane >> 4) << 1;                  // 0 or 2
    const bool lo   = lane < 16;

    const float sc = 2.0f / (float)SPAT;
    const float nx = ((float)w + 0.5f) * sc - 1.0f;      // own base voxel
    const float ny = ((float)h + 0.5f) * sc - 1.0f;
    const float nz = ((float)d + 0.5f) * sc - 1.0f;

    // ---- A (16x4) = theta zero-padded. Lanes 0-15: K=0,1; 16-31: K=2,3 ----
    const int   rs  = (row < 3) ? row : 0;
    const float ta0 = th[rs*4 + kb];
    const float ta1 = th[rs*4 + kb + 1];
    v2f Am;
    Am.x = (row < 3) ? ta0 : 0.0f;
    Am.y = (row < 3) ? ta1 : 0.0f;

    // ---- B (4x16): col n = base vector of lane (g*16+n)'s voxel.
    // VGPR0: lanes0-15 K=0 | lanes16-31 K=2; VGPR1: K=1 | K=3.
    const float nz_r = __shfl(nz, row, 32);        // col owners for B0 hi half
    const float nx_h = __shfl(nx, row + 16, 32);   // col owners for B1 lo half
    const float ny_h = __shfl(ny, row + 16, 32);
    v2f B0, B1;
    B0.x = lo ? nx   : nz_r;  B0.y = lo ? ny   : 1.0f;
    B1.x = lo ? nx_h : nz;    B1.y = lo ? ny_h : 1.0f;

    v8f cz = {0.f,0.f,0.f,0.f,0.f,0.f,0.f,0.f};
    v8f c0 = __builtin_amdgcn_wmma_f32_16x16x4_f32(false, Am, false, B0,
                                                   (short)0, cz, false, false);
    v8f c1 = __builtin_amdgcn_wmma_f32_16x16x4_f32(false, Am, false, B1,
                                                   (short)0, cz, false, false);

    // D rows 0..2 (gx,gy,gz) live in lanes 0-15, VGPR 0..2, col = lane.
    const float g0x = __shfl(c0[0], row, 32);
    const float g0y = __shfl(c0[1], row, 32);
    const float g0z = __shfl(c0[2], row, 32);
    const float g1x = __shfl(c1[0], row, 32);
    const float g1y = __shfl(c1[1], row, 32);
    const float g1z = __shfl(c1[2], row, 32);
    const float gx = lo ? g0x : g1x;
    const float gy = lo ? g0y : g1y;
    const float gz = lo ? g0z : g1z;

    // Affine map is linear in w: per-voxel step inside the quad.
    const float dgx = th[0] * sc;
    const float dgy = th[4] * sc;
    const float dgz = th[8] * sc;

    // ---- x-plane weights for the 4 voxels (only w-plane 48 is nonzero) ----
    float wx[WPT];
    #pragma unroll
    for (int i = 0; i < WPT; ++i) {
        float gxi = fmaf((float)i, dgx, gx);
        float ixi = ((gxi + 1.0f) * (float)SPAT - 1.0f) * 0.5f;
        float ff  = floorf(ixi);
        int   i0  = (int)ff;
        float fx  = ixi - ff;
        wx[i] = (i0 == SPAT/2) ? (1.0f - fx)
              : ((i0 == SPAT/2 - 1) ? fx : 0.0f);
    }
    const bool hit = (wx[0] != 0.0f) | (wx[1] != 0.0f)
                   | (wx[2] != 0.0f) | (wx[3] != 0.0f);

    float* op = out + (size_t)(b * 32 + s * NCH) * VOX + (size_t)q * WPT;

    if (!hit) {   // ~92% of quads: pure zero-fill with b128 NT stores
        const v4f z4 = {0.f, 0.f, 0.f, 0.f};
        #pragma unroll
        for (int c = 0; c < NCH; ++c)
            __builtin_nontemporal_store(z4, (v4f*)(op + (size_t)c * VOX));
        return;
    }

    // ---- hit path: bilinear (y,z) gather weights/offsets per voxel ----
    int   off[WPT][4];
    float wt[WPT][4];
    #pragma unroll
    for (int i = 0; i < WPT; ++i) {
        float gyi = fmaf((float)i, dgy, gy);
        float gzi = fmaf((float)i, dgz, gz);
        float iy  = ((gyi + 1.0f) * (float)SPAT - 1.0f) * 0.5f;
        float iz  = ((gzi + 1.0f) * (float)SPAT - 1.0f) * 0.5f;
        float fyf = floorf(iy); int iy0 = (int)fyf; float fy = iy - fyf;
        float fzf = floorf(iz); int iz0 = (int)fzf; float fz = iz - fzf;

        int y0 = iy0, y1 = iy0 + 1, z0 = iz0, z1 = iz0 + 1;
        bool vy0 = (y0 >= 0) & (y0 < SPAT);
        bool vy1 = (y1 >= 0) & (y1 < SPAT);
        bool vz0 = (z0 >= 0) & (z0 < SPAT);
        bool vz1 = (z1 >= 0) & (z1 < SPAT);

        float wxi = wx[i];
        wt[i][0] = wxi * (1.0f - fz) * (1.0f - fy) * ((vz0 && vy0) ? 1.0f : 0.0f);
        wt[i][1] = wxi * (1.0f - fz) * fy          * ((vz0 && vy1) ? 1.0f : 0.0f);
        wt[i][2] = wxi * fz          * (1.0f - fy) * ((vz1 && vy0) ? 1.0f : 0.0f);
        wt[i][3] = wxi * fz          * fy          * ((vz1 && vy1) ? 1.0f : 0.0f);

        int y0c = min(max(y0, 0), SPAT - 1);
        int y1c = min(max(y1, 0), SPAT - 1);
        int z0c = min(max(z0, 0), SPAT - 1);
        int z1c = min(max(z1, 0), SPAT - 1);
        off[i][0] = z0c * SPAT + y0c;
        off[i][1] = z0c * SPAT + y1c;
        off[i][2] = z1c * SPAT + y0c;
        off[i][3] = z1c * SPAT + y1c;
    }

    const float* xp = x + (size_t)(b * 32 + s * NCH) * XSL;
    #pragma unroll
    for (int c = 0; c < NCH; ++c) {
        const float* xc = xp + (size_t)c * XSL;     // x is 2.4 MB: L2-resident
        v4f v;
        #pragma unroll
        for (int i = 0; i < WPT; ++i) {
            v[i] = wt[i][0] * xc[off[i][0]] + wt[i][1] * xc[off[i][1]]
                 + wt[i][2] * xc[off[i][2]] + wt[i][3] * xc[off[i][3]];
        }
        __builtin_nontemporal_store(v, (v4f*)(op + (size_t)c * VOX));
    }
}

extern "C" void kernel_launch(void* const* d_in, const int* in_sizes, int n_in,
                              void* d_out, int out_size, void* d_ws, size_t ws_size,
                              hipStream_t stream) {
    (void)in_sizes; (void)n_in; (void)out_size; (void)d_ws; (void)ws_size;
    const float* x    = (const float*)d_in[0];   // [2,32,96,96] fp32
    const float* affs = (const float*)d_in[1];   // [2,2,4,4]    fp32
    float* out = (float*)d_out;                  // [2,32,96,96,96] fp32

    dim3 grid(NQ / THREADS, 4, 1);               // 864 x (slice*batch)
    skip_sample_kernel<<<grid, dim3(THREADS, 1, 1), 0, stream>>>(x, affs, out);
}